// ProposalCreator_44263932952806
// MI455X (gfx1250) — compile-verified
//
#include <hip/hip_runtime.h>
#include <math.h>

// ---------------------------------------------------------------------------
// ProposalCreator for MI455X (gfx1250, wave32, WGP w/ 320KB LDS)
//
// Pipeline (per image b, level l):
//   K1 pc_topk_sort : exact 4-pass 256-bin radix select of top-K scores
//                     (suffix-scan of bins on the matrix pipe via
//                      V_WMMA_F32_16X16X4_F32, exact for counts < 2^24),
//                     then in-LDS bitonic sort of the K survivors.
//   K2 pc_decode    : anchors computed analytically from index; box decode+clip.
//   K3 pc_nms       : greedy NMS, whole problem resident in LDS
//                     (12000 boxes = 187.5KB fits CDNA5's 320KB WGP LDS),
//                     monotone first-unsuppressed pointer over a bitmask.
//   K4 pc_final_topk: per-image bitonic top-2000 across 5 levels; -inf -> pad.
// ---------------------------------------------------------------------------

typedef float v2f __attribute__((ext_vector_type(2)));
typedef float v8f __attribute__((ext_vector_type(8)));

#define NLEV      5
#define NIMG      16
#define K_CAP     12000
#define TOPN_POST 2000
#define NMS_TH    0.7f
#define SORT_CAP  16384
#define NW_CAP    376   // ceil(12000/32)=375, padded

__constant__ int   c_N[NLEV]      = {196608, 49152, 12288, 3072, 768};
__constant__ int   c_K[NLEV]      = {12000, 12000, 12000, 3072, 768};
__constant__ int   c_S[NLEV]      = {16384, 16384, 16384, 4096, 1024};
__constant__ int   c_FW[NLEV]     = {256, 128, 64, 32, 16};
__constant__ float c_STRIDE[NLEV] = {4.f, 8.f, 16.f, 32.f, 64.f};
__constant__ float c_SIZE[NLEV]   = {32.f, 64.f, 128.f, 256.f, 512.f};
__constant__ float c_SQRTR[3]     = {0.70710678118654752f, 1.f, 1.41421356237309505f};

// monotone float -> u32 key (descending float == descending unsigned key)
static __device__ __forceinline__ unsigned fkey(float f){
  unsigned u = __float_as_uint(f);
  return (u & 0x80000000u) ? ~u : (u | 0x80000000u);
}
static __device__ __forceinline__ float unfkey(unsigned u){
  unsigned b = (u & 0x80000000u) ? (u & 0x7FFFFFFFu) : ~u;
  return __uint_as_float(b);
}
// pack key (hi) + ~idx (lo) so ties break toward the LOWEST index, like lax.top_k
static __device__ __forceinline__ unsigned long long packKI(unsigned key, unsigned idx){
  return ((unsigned long long)key << 32) | (unsigned long long)(~idx);
}

static __device__ void bitonic_sort_desc(unsigned long long* p, int S){
  for (int ksz = 2; ksz <= S; ksz <<= 1){
    for (int j = ksz >> 1; j > 0; j >>= 1){
      for (int i = (int)threadIdx.x; i < S; i += (int)blockDim.x){
        int ixj = i ^ j;
        if (ixj > i){
          unsigned long long a = p[i], b = p[ixj];
          bool up = ((i & ksz) == 0);
          bool sw = up ? (a < b) : (a > b);   // descending overall
          if (sw){ p[i] = b; p[ixj] = a; }
        }
      }
      __syncthreads();
    }
  }
}

// ---------------------------------------------------------------------------
// K1: exact top-K selection (sorted descending) per (level, image)
// ---------------------------------------------------------------------------
extern "C" __global__ void __launch_bounds__(1024)
pc_topk_sort(const float* p0, const float* p1, const float* p2,
             const float* p3, const float* p4,
             float* selScore, unsigned* selIdx)
{
  extern __shared__ unsigned char s_dyn[];
  unsigned long long* pairs = (unsigned long long*)s_dyn;   // SORT_CAP * 8B = 128KB
  __shared__ unsigned s_hist[256];
  __shared__ float    s_P[256];   // row-suffix matrix from WMMA
  __shared__ float    s_rs[16];   // row-total suffix
  __shared__ int      s_cntGt, s_cntEq;

  const int l = blockIdx.x, b = blockIdx.y;
  const int tid = (int)threadIdx.x, bdim = (int)blockDim.x;
  const int N = c_N[l], K = c_K[l], S = c_S[l];
  const float* pb;
  switch (l){ case 0: pb = p0; break; case 1: pb = p1; break; case 2: pb = p2; break;
              case 3: pb = p3; break; default: pb = p4; break; }
  pb += (size_t)b * (size_t)N;

  if (N > K){
    // ---- 4-pass radix refinement: find exact K-th largest key ----
    unsigned prefix = 0; int kq = K;
    for (int pass = 0; pass < 4; ++pass){
      const int shift = 24 - 8*pass;
      for (int i = tid; i < 256; i += bdim) s_hist[i] = 0u;
      __syncthreads();
      for (int i = tid; i < N; i += bdim){
        unsigned u = fkey(pb[i]);
        bool ok = (pass == 0);
        if (!ok){
          const int hs = shift + 8;
          ok = ((u >> hs) == (prefix >> hs));
        }
        if (ok) atomicAdd(&s_hist[(u >> shift) & 255u], 1u);
      }
      __syncthreads();
      // ---- 256-bin suffix scan on the matrix pipe (wave 0, EXEC all-ones) ----
      // H[r][c] = hist[16r+c].  P = H x B, B[k][n] = (k >= n)  =>
      // P[r][n] = sum_{c>=n} H[r][c]  (row suffix, diagonal inclusive).
      // suf(16r+n) = P[r][n] + sum_{r'>r} P[r'][0].  All counts < 2^24: exact f32.
      if (tid < 32){
        const int lane = tid, halfId = lane >> 4, mn = lane & 15;
        v8f acc = {0.f,0.f,0.f,0.f,0.f,0.f,0.f,0.f};
        #pragma unroll
        for (int q = 0; q < 4; ++q){              // K=16 as 4 chained k=4 WMMAs
          const int k0 = 4*q + 2*halfId;          // A/B VGPR wrap: lanes16-31 hold K+2
          v2f a, bb;
          a.x  = (float)s_hist[mn*16 + k0];
          a.y  = (float)s_hist[mn*16 + k0 + 1];
          bb.x = (k0     >= mn) ? 1.f : 0.f;
          bb.y = (k0 + 1 >= mn) ? 1.f : 0.f;
          acc = __builtin_amdgcn_wmma_f32_16x16x4_f32(false, a, false, bb,
                                                      (short)0, acc, false, false);
        }
        #pragma unroll
        for (int v = 0; v < 8; ++v)               // C/D layout: M = v + 8*half, N = lane&15
          s_P[(v + 8*halfId)*16 + mn] = acc[v];
      }
      __syncthreads();
      if (tid == 0){
        float run = 0.f;
        for (int r = 15; r >= 0; --r){ s_rs[r] = run; run += s_P[r*16]; }
      }
      __syncthreads();
      // every thread redundantly finds threshold bin t: suf[t] >= kq > suf[t+1]
      const float kqf = (float)kq;
      int t = 0; float sub = 0.f, prev = 0.f;
      for (int tt = 255; tt >= 0; --tt){
        float v = s_P[(tt>>4)*16 + (tt&15)] + s_rs[tt>>4];
        if (v >= kqf){ t = tt; sub = prev; break; }
        prev = v;
      }
      kq -= (int)(sub + 0.5f);
      prefix |= ((unsigned)t) << shift;
      __syncthreads();
    }
    // ---- compaction: strictly-greater all in; ==T fills remaining quota ----
    const unsigned T = prefix;
    const int cGt = K - kq;
    if (tid == 0){ s_cntGt = 0; s_cntEq = 0; }
    __syncthreads();
    for (int i = tid; i < N; i += bdim){
      unsigned u = fkey(pb[i]);
      if (u > T){
        int p = atomicAdd(&s_cntGt, 1);
        pairs[p] = packKI(u, (unsigned)i);
      } else if (u == T){
        int p = atomicAdd(&s_cntEq, 1);
        if (p < kq) pairs[cGt + p] = packKI(u, (unsigned)i);
      }
    }
    __syncthreads();
    for (int i = tid; i < S; i += bdim) if (i >= K) pairs[i] = 0ull;
    __syncthreads();
  } else {
    // small levels: sort everything
    for (int i = tid; i < S; i += bdim)
      pairs[i] = (i < N) ? packKI(fkey(pb[i]), (unsigned)i) : 0ull;
    __syncthreads();
  }

  bitonic_sort_desc(pairs, S);

  const size_t base = (size_t)(b*NLEV + l) * K_CAP;
  for (int i = tid; i < K; i += bdim){
    unsigned long long pr = pairs[i];
    selScore[base + i] = unfkey((unsigned)(pr >> 32));
    selIdx[base + i]   = ~((unsigned)(pr & 0xFFFFFFFFull));
  }
}

// ---------------------------------------------------------------------------
// K2: decode + clip boxes for selected anchors (anchors computed from index)
// ---------------------------------------------------------------------------
extern "C" __global__ void __launch_bounds__(256)
pc_decode(const float* r0, const float* r1, const float* r2,
          const float* r3, const float* r4,
          const float* img_info, const unsigned* selIdx, float4* boxes)
{
  const int l = blockIdx.y, b = blockIdx.z;
  const int i = (int)(blockIdx.x * blockDim.x + threadIdx.x);
  const int K = c_K[l], N = c_N[l];
  if (i >= K) return;
  const float* rg;
  switch (l){ case 0: rg = r0; break; case 1: rg = r1; break; case 2: rg = r2; break;
              case 3: rg = r3; break; default: rg = r4; break; }
  const size_t base = (size_t)(b*NLEV + l) * K_CAP;
  const unsigned idx = selIdx[base + i];
  const int ridx = (int)(idx % 3u);
  const int cell = (int)(idx / 3u);
  const int FW = c_FW[l];
  const int xi = cell % FW, yi = cell / FW;
  const float sr  = c_SQRTR[ridx];
  const float wsA = c_SIZE[l] / sr, hsA = c_SIZE[l] * sr;
  const float cxA = ((float)xi + 0.5f) * c_STRIDE[l];
  const float cyA = ((float)yi + 0.5f) * c_STRIDE[l];
  const float ax1 = cxA - wsA*0.5f, ay1 = cyA - hsA*0.5f;
  const float ax2 = cxA + wsA*0.5f, ay2 = cyA + hsA*0.5f;
  const float aw = ax2 - ax1, ah = ay2 - ay1;
  const float acx = ax1 + 0.5f*aw, acy = ay1 + 0.5f*ah;
  const float* d = rg + ((size_t)b * (size_t)N + (size_t)idx) * 4;
  const float cx = d[0]*aw + acx;
  const float cy = d[1]*ah + acy;
  const float w  = aw * expf(d[2]);
  const float h  = ah * expf(d[3]);
  const float Hh = img_info[2*b], Ww = img_info[2*b + 1];
  const float x1 = fminf(fmaxf(cx - 0.5f*w, 0.f), Ww);
  const float y1 = fminf(fmaxf(cy - 0.5f*h, 0.f), Hh);
  const float x2 = fminf(fmaxf(cx + 0.5f*w, 0.f), Ww);
  const float y2 = fminf(fmaxf(cy + 0.5f*h, 0.f), Hh);
  boxes[base + i] = make_float4(x1, y1, x2, y2);
}

// ---------------------------------------------------------------------------
// K3: greedy NMS, whole problem LDS-resident (CDNA5 320KB WGP LDS)
// ---------------------------------------------------------------------------
extern "C" __global__ void __launch_bounds__(1024)
pc_nms(const float4* boxes, const float* selScore, float4* keptBox, float* keptScore)
{
  extern __shared__ unsigned char s_dyn[];
  float4*   bx   = (float4*)s_dyn;                                        // 192000 B
  unsigned* sup  = (unsigned*)(s_dyn + (size_t)K_CAP*16);                 //   1504 B
  int*      keep = (int*)(s_dyn + (size_t)K_CAP*16 + (size_t)NW_CAP*4);   //   8000 B
  __shared__ int s_ptr, s_cur;

  const int l = blockIdx.x, b = blockIdx.y;
  const int tid = (int)threadIdx.x, bdim = (int)blockDim.x;
  const int K = c_K[l];
  const int NW = (K + 31) >> 5;
  const size_t base = (size_t)(b*NLEV + l) * K_CAP;

  for (int i = tid; i < K; i += bdim) bx[i] = boxes[base + i];
  for (int w = tid; w < NW; w += bdim){
    unsigned v = 0u;
    if (w == NW - 1 && (K & 31)) v = ~((1u << (K & 31)) - 1u);  // mark tail suppressed
    sup[w] = v;
  }
  if (tid == 0) s_ptr = 0;
  __syncthreads();

  for (int it = 0; it < TOPN_POST; ++it){
    if (tid == 0){
      // monotone pointer: bits never clear, so total scan work is O(K + iters)
      int p = s_ptr, cur = -1;
      int w = p >> 5;
      while (w < NW){
        unsigned avail = ~sup[w];
        if (w == (p >> 5)) avail &= (0xFFFFFFFFu << (p & 31));
        if (avail){ cur = (w << 5) + (__ffs(avail) - 1); break; }
        ++w;
      }
      s_cur = cur;
      keep[it] = cur;
      s_ptr = (cur >= 0) ? (cur + 1) : (NW << 5);
    }
    __syncthreads();
    const int cur = s_cur;
    if (cur >= 0){
      const float4 pbx = bx[cur];
      const float  pa  = (pbx.z - pbx.x) * (pbx.w - pbx.y);
      for (int j = cur + 1 + tid; j < K; j += bdim){
        if ((sup[j >> 5] >> (j & 31)) & 1u) continue;
        const float4 q = bx[j];
        const float x1 = fmaxf(pbx.x, q.x), y1 = fmaxf(pbx.y, q.y);
        const float x2 = fminf(pbx.z, q.z), y2 = fminf(pbx.w, q.w);
        const float inter = fmaxf(x2 - x1, 0.f) * fmaxf(y2 - y1, 0.f);
        const float qa = (q.z - q.x) * (q.w - q.y);
        const float iou = inter / (pa + qa - inter + 1e-9f);
        if (iou > NMS_TH) atomicOr(&sup[j >> 5], 1u << (j & 31));
      }
    }
    __syncthreads();
  }

  const size_t obase = (size_t)(b*NLEV + l) * TOPN_POST;
  for (int it = tid; it < TOPN_POST; it += bdim){
    const int kp = keep[it];
    float4 ob = make_float4(0.f, 0.f, 0.f, 0.f);
    float  os = -__builtin_inff();
    if (kp >= 0){ ob = bx[kp]; os = selScore[base + kp]; }
    keptBox[obase + it]   = ob;
    keptScore[obase + it] = os;
  }
}

// ---------------------------------------------------------------------------
// K4: per-image cross-level top-2000 + pad rows for -inf scores
// ---------------------------------------------------------------------------
extern "C" __global__ void __launch_bounds__(1024)
pc_final_topk(const float4* keptBox, const float* keptScore, float* out)
{
  extern __shared__ unsigned char s_dyn[];
  unsigned long long* pairs = (unsigned long long*)s_dyn;
  const int b = (int)blockIdx.x;
  const int tid = (int)threadIdx.x, bdim = (int)blockDim.x;
  const int M = NLEV * TOPN_POST;   // 10000
  const int S = SORT_CAP;           // 16384
  for (int i = tid; i < S; i += bdim){
    unsigned long long v = 0ull;
    if (i < M) v = packKI(fkey(keptScore[(size_t)b*M + i]), (unsigned)i);
    pairs[i] = v;
  }
  __syncthreads();
  bitonic_sort_desc(pairs, S);
  float4* out4 = (float4*)out;
  for (int j = tid; j < TOPN_POST; j += bdim){
    const unsigned long long pr = pairs[j];
    const float    s = unfkey((unsigned)(pr >> 32));
    const unsigned c = ~((unsigned)(pr & 0xFFFFFFFFull));
    float4 ob = make_float4(0.f, 0.f, 1.f, 1.f);   // reference pad row
    if (__builtin_isfinite(s)) ob = keptBox[(size_t)b*M + c];
    out4[(size_t)b*TOPN_POST + j] = ob;
  }
}

// ---------------------------------------------------------------------------
// Host launcher
// ---------------------------------------------------------------------------
extern "C" void kernel_launch(void* const* d_in, const int* in_sizes, int n_in,
                              void* d_out, int out_size, void* d_ws, size_t ws_size,
                              hipStream_t stream)
{
  (void)in_sizes; (void)n_in; (void)out_size; (void)ws_size;
  const float* p0 = (const float*)d_in[0];
  const float* p1 = (const float*)d_in[1];
  const float* p2 = (const float*)d_in[2];
  const float* p3 = (const float*)d_in[3];
  const float* p4 = (const float*)d_in[4];
  const float* r0 = (const float*)d_in[5];
  const float* r1 = (const float*)d_in[6];
  const float* r2 = (const float*)d_in[7];
  const float* r3 = (const float*)d_in[8];
  const float* r4 = (const float*)d_in[9];
  const float* img = (const float*)d_in[10];

  // workspace layout (all 16B aligned); total ~26.2 MB, L2-resident
  char* ws = (char*)d_ws;
  float*    selScore  = (float*)   (ws + 0);          // 80*12000*4  = 3,840,000
  unsigned* selIdx    = (unsigned*)(ws + 3840000);    // 80*12000*4  = 3,840,000
  float4*   boxes     = (float4*)  (ws + 7680000);    // 80*12000*16 = 15,360,000
  float4*   keptBox   = (float4*)  (ws + 23040000);   // 80*2000*16  = 2,560,000
  float*    keptScore = (float*)   (ws + 25600000);   // 80*2000*4   =   640,000

  const size_t smemSort = (size_t)SORT_CAP * 8ull;                                        // 128 KB
  const size_t smemNms  = (size_t)K_CAP*16 + (size_t)NW_CAP*4 + (size_t)TOPN_POST*4;      // ~197 KB (fits 320KB WGP LDS)

  pc_topk_sort<<<dim3(NLEV, NIMG), 1024, smemSort, stream>>>(p0, p1, p2, p3, p4,
                                                             selScore, selIdx);
  pc_decode<<<dim3((K_CAP + 255) / 256, NLEV, NIMG), 256, 0, stream>>>(r0, r1, r2, r3, r4,
                                                                       img, selIdx, boxes);
  pc_nms<<<dim3(NLEV, NIMG), 1024, smemNms, stream>>>(boxes, selScore, keptBox, keptScore);
  pc_final_topk<<<NIMG, 1024, smemSort, stream>>>(keptBox, keptScore, (float*)d_out);
}